// SimpleDecoder_85968065397366
// MI455X (gfx1250) — compile-verified
//
#include <hip/hip_runtime.h>
#include <math.h>

// CDNA5 wave32 WMMA fragment types for V_WMMA_F32_16X16X4_F32:
//   A (16x4 f32)  = 64 vals / 32 lanes = 2 VGPRs -> v2f
//   B (4x16 f32)  = 64 vals / 32 lanes = 2 VGPRs -> v2f
//   C/D (16x16)   = 256 vals / 32 lanes = 8 VGPRs -> v8f
typedef __attribute__((ext_vector_type(2))) float v2f;
typedef __attribute__((ext_vector_type(8))) float v8f;

__device__ __forceinline__ float gelu_exact(float x) {
    // jax.nn.gelu(approximate=False): 0.5*x*(1+erf(x/sqrt(2)))
    return 0.5f * x * (1.0f + erff(x * 0.70710678118654752f));
}

// Zero the 1024-float padding row used by A-fragment lanes whose batch row
// doesn't exist (rows 8..15 of the padded 16-row M dimension).
__global__ __launch_bounds__(256) void zpad_kernel(float* __restrict__ zpad) {
    zpad[threadIdx.x]       = 0.0f;
    zpad[threadIdx.x + 256] = 0.0f;
    zpad[threadIdx.x + 512] = 0.0f;
    zpad[threadIdx.x + 768] = 0.0f;
}

// One layer of  out = act(act_in @ W^T + bias).
//   act_in : 8 x IN  row-major (batch rows 0..7)
//   W      : OUTN x IN row-major  -> B[k][n] = W[n*IN + k] (no transpose needed)
//   act_out: 8 x OUTN row-major
// Each wave owns one 16-wide N tile; K loop in steps of 8 via two 4-deep f32 WMMAs.
// Lanes with l15 >= 8 read A from zpad (all zeros) -> no EXEC manipulation in
// the inner loop, pure load->wmma stream.
template<bool FINAL>
__global__ __launch_bounds__(256) void layer_kernel(
    const float* __restrict__ act_in,
    const float* __restrict__ W,
    const float* __restrict__ bias,
    const float* __restrict__ zpad,     // >= IN zeroed floats
    float* __restrict__ act_out,
    int IN, int OUTN)
{
    const int wave = threadIdx.x >> 5;
    const int lane = threadIdx.x & 31;
    const int l15  = lane & 15;
    const int half = lane >> 4;                 // 0 -> K={0,1}, 1 -> K={2,3} within a k-step

    const int tile = blockIdx.x * 8 + wave;     // 8 waves per 256-thread block
    const int nt   = tile * 16;
    if (nt >= OUTN) return;                     // wave-uniform exit

    const int n = nt + l15;                     // this lane's output column
    const float* wrow = W + (size_t)n * IN + 2 * half;
    // Per-lane pointer select done ONCE: padding rows read zeros from zpad.
    const float* arow = ((l15 < 8) ? (act_in + (size_t)l15 * IN) : zpad) + 2 * half;

    v8f c0 = {}; v8f c1 = {};                   // two accumulators -> break WMMA RAW chain
    for (int kt = 0; kt < IN; kt += 8) {
        v2f a0 = *(const v2f*)(arow + kt);
        v2f a1 = *(const v2f*)(arow + kt + 4);
        v2f b0 = *(const v2f*)(wrow + kt);
        v2f b1 = *(const v2f*)(wrow + kt + 4);
        c0 = __builtin_amdgcn_wmma_f32_16x16x4_f32(false, a0, false, b0, (short)0, c0, false, false);
        c1 = __builtin_amdgcn_wmma_f32_16x16x4_f32(false, a1, false, b1, (short)0, c1, false, false);
    }
    c0 += c1;

    const float bv = bias[n];
    if (lane < 16) {                            // lanes 0..15: VGPR i holds row M=i (batch i)
        #pragma unroll
        for (int i = 0; i < 8; ++i) {
            float v = c0[i] + bv;
            if (FINAL) v = 1.0f / (1.0f + expf(-v));   // sigmoid
            else       v = gelu_exact(v);
            act_out[(size_t)i * OUTN + n] = v;
        }
    }
}

// Broadcast: out_flat[b, f] = o[b, f / 12306]  ->  1536 contiguous runs of 12306 floats.
// One block per run. Run base byte offset = run * 49224; 49224 % 16 == 8, so odd runs
// need a 2-float head to reach 16B alignment, then b128 stores, then a <=2-float tail.
__global__ __launch_bounds__(256) void bcast_kernel(
    const float* __restrict__ o, float* __restrict__ out)
{
    const int run = blockIdx.x;                 // 0..1535  (= b*192 + j)
    const float v = o[run];                     // uniform -> s_load, broadcast
    float* p = out + (size_t)run * 12306u;

    const int head  = (run & 1) ? 2 : 0;        // align to 16B
    const int count = 12306 - head;
    const int quads = count >> 2;               // 3076 either way
    const int tail  = count & 3;                // 2 (even run) or 0 (odd run)

    if (threadIdx.x == 0 && head) { p[0] = v; p[1] = v; }

    float4* q = (float4*)(p + head);
    const float4 vv = make_float4(v, v, v, v);
    for (int i = threadIdx.x; i < quads; i += 256)
        q[i] = vv;                              // global_store_b128

    if (threadIdx.x == 1 && tail) {
        p[head + 4 * quads + 0] = v;
        p[head + 4 * quads + 1] = v;
    }
}

extern "C" void kernel_launch(void* const* d_in, const int* in_sizes, int n_in,
                              void* d_out, int out_size, void* d_ws, size_t ws_size,
                              hipStream_t stream) {
    const float* x  = (const float*)d_in[0];    // (8,1,256) == 8x256
    const float* W1 = (const float*)d_in[1];    // 512x256
    const float* b1 = (const float*)d_in[2];
    const float* W2 = (const float*)d_in[3];    // 1024x512
    const float* b2 = (const float*)d_in[4];
    const float* W3 = (const float*)d_in[5];    // 512x1024
    const float* b3 = (const float*)d_in[6];
    const float* W4 = (const float*)d_in[7];    // 192x512
    const float* b4 = (const float*)d_in[8];
    float* out = (float*)d_out;

    // Workspace layout (floats): o[2048] (1536 used) | zpad[1024] | a1[8x512] | a2[8x1024] | a3[8x512]
    float* w    = (float*)d_ws;
    float* o_   = w;
    float* zpad = w + 2048;
    float* a1   = zpad + 1024;
    float* a2   = a1 + 8 * 512;
    float* a3   = a2 + 8 * 1024;

    zpad_kernel<<<dim3(1), dim3(256), 0, stream>>>(zpad);

    // N tiles: L1=32, L2=64, L3=32, L4=12; 8 waves (tiles) per block.
    layer_kernel<false><<<dim3(4), dim3(256), 0, stream>>>(x,  W1, b1, zpad, a1, 256,  512);
    layer_kernel<false><<<dim3(8), dim3(256), 0, stream>>>(a1, W2, b2, zpad, a2, 512,  1024);
    layer_kernel<false><<<dim3(4), dim3(256), 0, stream>>>(a2, W3, b3, zpad, a3, 1024, 512);
    layer_kernel<true ><<<dim3(2), dim3(256), 0, stream>>>(a3, W4, b4, zpad, o_, 512,  192);

    // 8*192 = 1536 runs of 12306 floats each -> 75.6 MB of stores (the real cost).
    bcast_kernel<<<dim3(1536), dim3(256), 0, stream>>>(o_, out);
}